// EEGvae_87514253624051
// MI455X (gfx1250) — compile-verified
//
#include <hip/hip_runtime.h>
#include <hip/hip_bf16.h>
#include <math.h>

typedef __attribute__((ext_vector_type(2))) float v2f;
typedef __attribute__((ext_vector_type(8))) float v8f;

#define WMMA_F32(a, b, c) \
  __builtin_amdgcn_wmma_f32_16x16x4_f32(false, (a), false, (b), (short)0, (c), false, false)

#define NB 65536
#define CH 12
#define LRAW 63
#define FLEN 7
#define VLEN 57
#define FG_IN 168
#define FG_H 35
#define T1 18   // conv1 output length
#define T2 5    // conv2 output length
#define OC1 16
#define OC2 32

// CDNA5 has native V_TANH_F32 / V_EXP_F32 trans ops; use them when the
// builtin exists (gfx1250), fall back to libm otherwise so compile never breaks.
__device__ __forceinline__ float fast_tanh(float x) {
#if __has_builtin(__builtin_amdgcn_tanh_f32)
  return __builtin_amdgcn_tanh_f32(x);
#elif __has_builtin(__builtin_amdgcn_tanhf)
  return __builtin_amdgcn_tanhf(x);
#else
  return tanhf(x);
#endif
}

__device__ __forceinline__ float fast_exp(float x) {
#if __has_builtin(__builtin_amdgcn_exp2f)
  return __builtin_amdgcn_exp2f(x * 1.44269504088896340736f);
#else
  return expf(x);
#endif
}

// ---------------------------------------------------------------------------
// Kernel 1: filter-gen (WMMA f32 GEMM) -> adaptive 7-tap filter -> conv1
// (WMMA f32 GEMM, im2col per t) -> a1 preactivations + BN1 per-block partials
// One wave (32 threads) per 16-sample tile.
// ---------------------------------------------------------------------------
__global__ __launch_bounds__(32) void eegvae_k1(
    const float* __restrict__ raw, const float* __restrict__ eegf,
    const float* __restrict__ fg_w1, const float* __restrict__ fg_b1,
    const float* __restrict__ fg_w2, const float* __restrict__ fg_b2,
    const float* __restrict__ enc_w1, const float* __restrict__ enc_b1,
    float* __restrict__ a1, float* __restrict__ part1)
{
  __shared__ __align__(16) float hbuf[16][48];
  __shared__ __align__(16) float filtL[16][8];
  __shared__ __align__(16) float eegL[16][VLEN][CH];   // [s][u][ic], ic-contig

  const int lane = threadIdx.x;
  const int b0   = blockIdx.x * 16;
  const int m    = lane & 15;
  const int hi   = lane >> 4;
  const int koff = hi * 2;

  // ---- GEMM1: H[16x35(pad48)] = eegf[16x168] @ fg_w1^T, then tanh ----
  v8f acc0 = {0.f,0.f,0.f,0.f,0.f,0.f,0.f,0.f};
  v8f acc1 = acc0, acc2 = acc0;
  {
    const float* arow = eegf + (size_t)(b0 + m) * FG_IN + koff;
    for (int j = 0; j < FG_IN / 4; ++j) {
      v2f a = *(const v2f*)(arow + 4 * j);
      {   // N-tile 0: cols 0..15
        const float* wr = fg_w1 + (size_t)m * FG_IN + 4 * j + koff;
        v2f b; b.x = wr[0]; b.y = wr[1];
        acc0 = WMMA_F32(a, b, acc0);
      }
      {   // N-tile 1: cols 16..31
        const float* wr = fg_w1 + (size_t)(16 + m) * FG_IN + 4 * j + koff;
        v2f b; b.x = wr[0]; b.y = wr[1];
        acc1 = WMMA_F32(a, b, acc1);
      }
      {   // N-tile 2: cols 32..47 (rows >=35 clamped; junk cols unused)
        int n = 32 + m; int nr = n < FG_H ? n : FG_H - 1;
        const float* wr = fg_w1 + (size_t)nr * FG_IN + 4 * j + koff;
        v2f b; b.x = wr[0]; b.y = wr[1];
        acc2 = WMMA_F32(a, b, acc2);
      }
    }
  }
  {
    v8f accs[3] = {acc0, acc1, acc2};
#pragma unroll
    for (int nt = 0; nt < 3; ++nt) {
      int n = nt * 16 + m;
      float bias = (n < FG_H) ? fg_b1[n] : 0.f;
#pragma unroll
      for (int r = 0; r < 8; ++r) {
        int s = r + 8 * hi;                 // D layout: M = r + 8*(lane>=16)
        hbuf[s][n] = fast_tanh(accs[nt][r] + bias);
      }
    }
  }
  __syncthreads();

  // ---- GEMM2 (tiny, scalar): filt[16x7] = tanh(H @ fg_w2^T + b2) ----
  {
    int s = m;
    int kb = hi ? 4 : 0, ke = hi ? FLEN : 4;
    for (int k = kb; k < ke; ++k) {
      float acc = fg_b2[k];
      for (int n = 0; n < FG_H; ++n) acc += hbuf[s][n] * fg_w2[k * FG_H + n];
      filtL[s][k] = fast_tanh(acc);
    }
  }
  __syncthreads();

  // ---- adaptive depthwise filter: eeg[s][u][ic] (valid correlation) ----
  {
    const float* rawB = raw + (size_t)b0 * (CH * LRAW);
    for (int idx = lane; idx < 16 * VLEN; idx += 32) {
      int s = idx / VLEN, u = idx - s * VLEN;
      const float* f = filtL[s];
      for (int ic = 0; ic < CH; ++ic) {
        const float* rp = rawB + (size_t)s * (CH * LRAW) + ic * LRAW + u;
        float v = 0.f;
#pragma unroll
        for (int k = 0; k < FLEN; ++k) v += rp[k] * f[k];
        eegL[s][u][ic] = v;
      }
    }
  }
  __syncthreads();

  // ---- conv1 as WMMA GEMM per t: K = 72 ordered kidx = kw*12 + ic ----
  v2f Bf[18];
#pragma unroll
  for (int j = 0; j < 18; ++j) {
    int kidx = 4 * j + koff;
    int ic = kidx % CH, kw = kidx / CH;     // pair never crosses kw boundary
    const float* w = enc_w1 + m * (CH * 6) + ic * 6 + kw;
    Bf[j].x = w[0]; Bf[j].y = w[6];
  }
  float bsum = 0.f, bssq = 0.f;
  const float bias1 = enc_b1[m];
  for (int t = 0; t < T1; ++t) {
    v8f d = {0.f,0.f,0.f,0.f,0.f,0.f,0.f,0.f};
#pragma unroll
    for (int j = 0; j < 18; ++j) {
      int kidx = 4 * j + koff;
      int ic = kidx % CH, kw = kidx / CH;
      v2f a = *(const v2f*)&eegL[m][3 * t + kw][ic];   // ic even -> aligned
      d = WMMA_F32(a, Bf[j], d);
    }
#pragma unroll
    for (int r = 0; r < 8; ++r) {
      int s = r + 8 * hi;
      float v = d[r] + bias1;
      a1[(size_t)(b0 + s) * (T1 * OC1) + t * OC1 + m] = v;   // [b][t][oc]
      bsum += v; bssq += v * v;
    }
  }
  bsum += __shfl_xor(bsum, 16, 32);
  bssq += __shfl_xor(bssq, 16, 32);
  if (lane < 16) {
    part1[(size_t)blockIdx.x * 32 + lane]      = bsum;
    part1[(size_t)blockIdx.x * 32 + 16 + lane] = bssq;
  }
}

// ---------------------------------------------------------------------------
// BN finalize: deterministic fixed-order tree reduce of per-block partials.
// part layout: [blk][2*nch]  (first nch = sums, next nch = sumsq)
// ---------------------------------------------------------------------------
__global__ __launch_bounds__(256) void eegvae_kbn(
    const float* __restrict__ part, int nb, int nch,
    const float* __restrict__ g, const float* __restrict__ b,
    float* __restrict__ bnS, float* __restrict__ bnB, float invCount)
{
  __shared__ float red[64][8];
  __shared__ float tot[64];
  const int t = threadIdx.x;
  const int ns = 2 * nch;            // 32 or 64
  const int nsub = 256 / ns;         // 8 or 4
  const int s = t % ns, sub = t / ns;
  float acc = 0.f;
  for (int p = sub; p < nb; p += nsub) acc += part[(size_t)p * ns + s];
  red[s][sub] = acc;
  __syncthreads();
  if (t < ns) {
    float a = 0.f;
    for (int q = 0; q < nsub; ++q) a += red[t][q];
    tot[t] = a;
  }
  __syncthreads();
  if (t < nch) {
    float mean = tot[t] * invCount;
    float var  = tot[nch + t] * invCount - mean * mean;
    float sc   = g[t] * rsqrtf(var + 1e-5f);
    bnS[t] = sc;
    bnB[t] = b[t] - mean * sc;
  }
}

// ---------------------------------------------------------------------------
// Kernel 3: h1 = tanh(bn1(a1)) applied inline while building A-fragments;
// conv2 as WMMA GEMM (K = 96, kidx = kw*16 + c1). a2 + BN2 partials.
// ---------------------------------------------------------------------------
__global__ __launch_bounds__(32) void eegvae_k3(
    const float* __restrict__ a1,
    const float* __restrict__ enc_w2, const float* __restrict__ enc_b2,
    const float* __restrict__ bn1S, const float* __restrict__ bn1B,
    float* __restrict__ a2, float* __restrict__ part2)
{
  __shared__ float sS[16], sB[16];
  const int lane = threadIdx.x;
  const int b0 = blockIdx.x * 16;
  if (lane < 16) { sS[lane] = bn1S[lane]; sB[lane] = bn1B[lane]; }
  __syncthreads();
  const int m = lane & 15, hi = lane >> 4, koff = hi * 2;

  for (int nt = 0; nt < 2; ++nt) {
    const int oc = nt * 16 + m;
    v2f Bf[24];
#pragma unroll
    for (int j = 0; j < 24; ++j) {
      int kidx = 4 * j + koff;
      int c1 = kidx & 15, kw = kidx >> 4;
      const float* w = enc_w2 + oc * (OC1 * 6) + c1 * 6 + kw;
      Bf[j].x = w[0]; Bf[j].y = w[6];
    }
    const float bias = enc_b2[oc];
    float bsum = 0.f, bssq = 0.f;
    for (int t2 = 0; t2 < T2; ++t2) {
      v8f d = {0.f,0.f,0.f,0.f,0.f,0.f,0.f,0.f};
#pragma unroll
      for (int j = 0; j < 24; ++j) {
        int kidx = 4 * j + koff;
        int c1 = kidx & 15, kw = kidx >> 4;
        const float* ap = a1 + (size_t)(b0 + m) * (T1 * OC1) + (3 * t2 + kw) * OC1 + c1;
        v2f r2 = *(const v2f*)ap;                      // c1 even -> aligned
        v2f a;
        a.x = fast_tanh(r2.x * sS[c1]     + sB[c1]);
        a.y = fast_tanh(r2.y * sS[c1 + 1] + sB[c1 + 1]);
        d = WMMA_F32(a, Bf[j], d);
      }
#pragma unroll
      for (int r = 0; r < 8; ++r) {
        int s = r + 8 * hi;
        float v = d[r] + bias;
        a2[(size_t)(b0 + s) * (OC2 * T2) + oc * T2 + t2] = v;   // [b][oc][t]
        bsum += v; bssq += v * v;
      }
    }
    bsum += __shfl_xor(bsum, 16, 32);
    bssq += __shfl_xor(bssq, 16, 32);
    if (lane < 16) {
      part2[(size_t)blockIdx.x * 64 + nt * 16 + lane]      = bsum;
      part2[(size_t)blockIdx.x * 64 + 32 + nt * 16 + lane] = bssq;
    }
  }
}

// ---------------------------------------------------------------------------
// Kernel 5: bn2+tanh -> mu/logvar heads -> reparameterize -> decoder -> out
// One sample per thread; a2 stored [b][oc][t] == reference h flatten order.
// ---------------------------------------------------------------------------
__global__ __launch_bounds__(256) void eegvae_k5(
    const float* __restrict__ a2, const float* __restrict__ eps,
    const float* __restrict__ bn2S, const float* __restrict__ bn2B,
    const float* __restrict__ mu_w, const float* __restrict__ mu_b,
    const float* __restrict__ lv_w, const float* __restrict__ lv_b,
    const float* __restrict__ de_w1, const float* __restrict__ de_b1,
    const float* __restrict__ de_w2, const float* __restrict__ de_b2,
    float* __restrict__ out, int batch)
{
  __shared__ float s2[32], sb2[32], wmu[320], wlv[320];
  __shared__ float w1d[20], b1d[10], w2d[500], b2d[50];
  const int t = threadIdx.x;
  for (int i = t; i < 32;  i += 256) { s2[i] = bn2S[i]; sb2[i] = bn2B[i]; }
  for (int i = t; i < 320; i += 256) { wmu[i] = mu_w[i]; wlv[i] = lv_w[i]; }
  for (int i = t; i < 20;  i += 256) w1d[i] = de_w1[i];
  for (int i = t; i < 10;  i += 256) b1d[i] = de_b1[i];
  for (int i = t; i < 500; i += 256) w2d[i] = de_w2[i];
  for (int i = t; i < 50;  i += 256) b2d[i] = de_b2[i];
  __syncthreads();

  const int b = blockIdx.x * 256 + t;
  const float* hv = a2 + (size_t)b * 160;
  float mu0 = mu_b[0], mu1 = mu_b[1], lv0 = lv_b[0], lv1 = lv_b[1];
  int i = 0;
  for (int oc = 0; oc < 32; ++oc) {
    float sc = s2[oc], sh = sb2[oc];
#pragma unroll
    for (int tt = 0; tt < 5; ++tt, ++i) {
      float h = fast_tanh(hv[i] * sc + sh);
      mu0 += h * wmu[i];       mu1 += h * wmu[160 + i];
      lv0 += h * wlv[i];       lv1 += h * wlv[160 + i];
    }
  }
  float z0 = mu0 + eps[(size_t)b * 2 + 0] * fast_exp(0.5f * lv0);
  float z1 = mu1 + eps[(size_t)b * 2 + 1] * fast_exp(0.5f * lv1);
  float dv[10];
#pragma unroll
  for (int j = 0; j < 10; ++j)
    dv[j] = fast_tanh(z0 * w1d[2 * j] + z1 * w1d[2 * j + 1] + b1d[j]);
  float* o = out + (size_t)b * 50;
  for (int k = 0; k < 50; ++k) {
    float acc = b2d[k];
#pragma unroll
    for (int j = 0; j < 10; ++j) acc += dv[j] * w2d[k * 10 + j];
    o[k] = fast_tanh(acc);
  }
  float* mup = out + (size_t)batch * 50;
  mup[(size_t)b * 2 + 0] = mu0; mup[(size_t)b * 2 + 1] = mu1;
  float* lvp = mup + (size_t)batch * 2;
  lvp[(size_t)b * 2 + 0] = lv0; lvp[(size_t)b * 2 + 1] = lv1;
}

// ---------------------------------------------------------------------------
extern "C" void kernel_launch(void* const* d_in, const int* in_sizes, int n_in,
                              void* d_out, int out_size, void* d_ws, size_t ws_size,
                              hipStream_t stream) {
  const float* raw    = (const float*)d_in[0];
  const float* eegf   = (const float*)d_in[1];
  const float* eps    = (const float*)d_in[2];
  const float* fg_w1  = (const float*)d_in[3];
  const float* fg_b1  = (const float*)d_in[4];
  const float* fg_w2  = (const float*)d_in[5];
  const float* fg_b2  = (const float*)d_in[6];
  const float* enc_w1 = (const float*)d_in[7];
  const float* enc_b1 = (const float*)d_in[8];
  const float* bn1_g  = (const float*)d_in[9];
  const float* bn1_b  = (const float*)d_in[10];
  const float* enc_w2 = (const float*)d_in[11];
  const float* enc_b2 = (const float*)d_in[12];
  const float* bn2_g  = (const float*)d_in[13];
  const float* bn2_b  = (const float*)d_in[14];
  const float* mu_w   = (const float*)d_in[15];
  const float* mu_b   = (const float*)d_in[16];
  const float* lv_w   = (const float*)d_in[17];
  const float* lv_b   = (const float*)d_in[18];
  const float* de_w1  = (const float*)d_in[19];
  const float* de_b1  = (const float*)d_in[20];
  const float* de_w2  = (const float*)d_in[21];
  const float* de_b2  = (const float*)d_in[22];
  float* out = (float*)d_out;

  const int batch = in_sizes[0] / (CH * LRAW);   // 65536
  const int ntile = batch / 16;                  // 4096 wave-tiles

  // workspace carve-up (floats)
  float* ws    = (float*)d_ws;
  float* a1    = ws;                                     // batch*288
  float* a2    = a1 + (size_t)batch * (T1 * OC1);        // batch*160
  float* part1 = a2 + (size_t)batch * (OC2 * T2);        // ntile*32
  float* part2 = part1 + (size_t)ntile * 32;             // ntile*64
  float* bn1S  = part2 + (size_t)ntile * 64;             // 16
  float* bn1B  = bn1S + 16;                              // 16
  float* bn2S  = bn1B + 16;                              // 32
  float* bn2B  = bn2S + 32;                              // 32

  eegvae_k1<<<ntile, 32, 0, stream>>>(raw, eegf, fg_w1, fg_b1, fg_w2, fg_b2,
                                      enc_w1, enc_b1, a1, part1);
  eegvae_kbn<<<1, 256, 0, stream>>>(part1, ntile, OC1, bn1_g, bn1_b,
                                    bn1S, bn1B, 1.0f / ((float)batch * T1));
  eegvae_k3<<<ntile, 32, 0, stream>>>(a1, enc_w2, enc_b2, bn1S, bn1B, a2, part2);
  eegvae_kbn<<<1, 256, 0, stream>>>(part2, ntile, OC2, bn2_g, bn2_b,
                                    bn2S, bn2B, 1.0f / ((float)batch * T2));
  eegvae_k5<<<batch / 256, 256, 0, stream>>>(a2, eps, bn2S, bn2B,
                                             mu_w, mu_b, lv_w, lv_b,
                                             de_w1, de_b1, de_w2, de_b2,
                                             out, batch);
}